// InverseWaveletPacketTransform_74715251081586
// MI455X (gfx1250) — compile-verified
//
#include <hip/hip_runtime.h>

typedef __attribute__((ext_vector_type(2))) float v2f;
typedef __attribute__((ext_vector_type(4))) float v4f;
typedef __attribute__((ext_vector_type(8))) float v8f;

#define NODE_STRIDE 16384
#define PLANE_ELEMS (8 * 16384) /* 131072 elems per (b,c) plane, in and out */

// Inverse Haar wavelet-packet transform, level 3, as a signed-Hadamard WMMA.
// out[plane, 8j + r] = 2^-1.5 * sum_k sign(r,k) * coeffs[plane, invgray(k), j]
// sign(r,k) = (-1)^{ r2*k0 + r1*k1 + r0*k2 },  invgray(k) = k ^ (k>>1) ^ (k>>2)
__global__ __launch_bounds__(256) void iwpt_haar3_wmma(
    const float* __restrict__ coeffs, float* __restrict__ out) {
  const int lane = threadIdx.x & 31;
  const int wave = (blockIdx.x * blockDim.x + threadIdx.x) >> 5;

  const int base_tile = wave * 4;          // 4 tiles of 32 j-values per wave
  const int plane     = base_tile >> 9;    // 512 tiles per plane
  const int j0_base   = (base_tile & 511) * 32;

  const int hi = lane >> 4;                // which half of the wave
  const int n  = lane & 15;                // column (B/D) or row (A) index
  const float SCALE = 0.35355339059327378f; // 2^-1.5

  // ---- A matrix: blockdiag(H8, H8), 16x16, split into four 16x4 K-slices.
  // A(m,col) lives in lane (m + 16*(col_local>=2)), vgpr (col_local&1).
  v2f aslab[4];
  int ld_off[4][2];  // per-lane global element offsets for the B loads
  #pragma unroll
  for (int q = 0; q < 4; ++q) {
    #pragma unroll
    for (int v = 0; v < 2; ++v) {
      const int col = 4 * q + 2 * hi + v;  // 0..15 row index rho of B / col of A
      // A value for row m=n, column col of blockdiag(H8,H8)
      float val = 0.0f;
      if ((n >= 8) == (col >= 8)) {
        const int r = n & 7, k = col & 7;
        const int rb = ((r & 1) << 2) | (r & 2) | ((r >> 2) & 1); // bitrev3(r)
        val = (__popc(rb & k) & 1) ? -SCALE : SCALE;
      }
      if (v == 0) aslab[q].x = val; else aslab[q].y = val;
      // B(rho, n): rho<8 -> coeffs[P[rho], j0+n]; rho>=8 -> coeffs[P[rho-8], j0+16+n]
      const int kk   = col & 7;
      const int node = kk ^ (kk >> 1) ^ (kk >> 2);  // inverse Gray permutation
      ld_off[q][v] = node * NODE_STRIDE + ((col >= 8) ? 16 : 0);
    }
  }

  const float* __restrict__ src = coeffs + (size_t)plane * PLANE_ELEMS;
  float* __restrict__ dstbase   = out    + (size_t)plane * PLANE_ELEMS;

  #pragma unroll
  for (int t = 0; t < 4; ++t) {
    const int j0 = j0_base + 32 * t;

    // Gather B: 8 coalesced b32 loads (each a contiguous 64B half-wave segment)
    v2f b[4];
    #pragma unroll
    for (int q = 0; q < 4; ++q) {
      b[q].x = src[ld_off[q][0] + j0 + n];
      b[q].y = src[ld_off[q][1] + j0 + n];
    }

    // D = blockdiag(H8,H8) x B, chained over K=16 (4 x K=4 WMMAs)
    v8f c = {};
    c = __builtin_amdgcn_wmma_f32_16x16x4_f32(false, aslab[0], false, b[0],
                                              (short)0, c, false, false);
    c = __builtin_amdgcn_wmma_f32_16x16x4_f32(false, aslab[1], false, b[1],
                                              (short)0, c, false, false);
    c = __builtin_amdgcn_wmma_f32_16x16x4_f32(false, aslab[2], false, b[2],
                                              (short)0, c, false, false);
    c = __builtin_amdgcn_wmma_f32_16x16x4_f32(false, aslab[3], false, b[3],
                                              (short)0, c, false, false);

    // Every lane L owns out[8*(j0+L) + 0..7]: two aligned b128 stores,
    // wave covers a contiguous 1 KiB block.
    float* dst = dstbase + (size_t)(j0 + lane) * 8;
    v4f lo, hi4;
    lo[0] = c[0]; lo[1] = c[1]; lo[2] = c[2]; lo[3] = c[3];
    hi4[0] = c[4]; hi4[1] = c[5]; hi4[2] = c[6]; hi4[3] = c[7];
    *(v4f*)(dst)     = lo;
    *(v4f*)(dst + 4) = hi4;
  }
}

extern "C" void kernel_launch(void* const* d_in, const int* in_sizes, int n_in,
                              void* d_out, int out_size, void* d_ws, size_t ws_size,
                              hipStream_t stream) {
  const float* coeffs = (const float*)d_in[0];
  float* out = (float*)d_out;
  // 256 planes * 512 tiles = 131072 tiles; 8 waves/block * 4 tiles/wave
  // -> 4096 blocks of 256 threads, exact cover, no tail.
  iwpt_haar3_wmma<<<4096, 256, 0, stream>>>(coeffs, out);
}